// MemoryImageUpdater_15393162789070
// MI455X (gfx1250) — compile-verified
//
#include <hip/hip_runtime.h>
#include <stdint.h>

// ---------------- problem constants (match reference) ----------------
constexpr int    NB    = 16;          // batch
constexpr int    NC    = 3;           // image channels
constexpr int    NCLS  = 8;           // classes
constexpr int    HW    = 512 * 512;   // pixels per image plane
constexpr int    KEEP  = 65536;       // ceil(0.25 * HW)
constexpr int    BINS  = 4096;
constexpr int    TILE  = 1024;        // pixels per block
constexpr float  DECAY = 0.9f;
constexpr float  TAU   = 0.05f;

// d_out flat layout (floats), reference tuple order
constexpr size_t OFF_MEMIMG = 0;
constexpr size_t OFF_MEMST  = (size_t)NB * NC * HW;                // 12,582,912
constexpr size_t OFF_SCORE  = OFF_MEMST + (size_t)NB * NC * HW;    // 25,165,824
constexpr size_t OFF_IMP    = OFF_SCORE + (size_t)NB * HW;
constexpr size_t OFF_WMASK  = OFF_IMP   + (size_t)NB * HW;
constexpr size_t OFF_OMASK  = OFF_WMASK + (size_t)NB * HW;

// d_ws layout (ints)
constexpr int WS_HIST  = 0;                    // NB * BINS
constexpr int WS_THR   = NB * BINS;            // NB
constexpr int WS_QUO   = WS_THR + NB;          // NB
constexpr int WS_BCNT  = WS_QUO + NB;          // NB * 256
constexpr int WS_TOTAL = WS_BCNT + NB * 256;   // 69,664 ints (~272 KB)

typedef unsigned int u32x4 __attribute__((ext_vector_type(4)));
typedef int          i32x4 __attribute__((ext_vector_type(4)));
typedef int          i32x8 __attribute__((ext_vector_type(8)));

#if defined(__AMDGCN__) && __has_builtin(__builtin_amdgcn_tensor_load_to_lds)
#define USE_TDM 1
#else
#define USE_TDM 0
#endif

__device__ __forceinline__ int bin_of(float x) {
    int b = (int)(x * (float)BINS);
    return b > (BINS - 1) ? (BINS - 1) : (b < 0 ? 0 : b);
}

__device__ __forceinline__ void wait_tensorcnt0() {
#if defined(__AMDGCN__) && __has_builtin(__builtin_amdgcn_s_wait_tensorcnt)
    __builtin_amdgcn_s_wait_tensorcnt(0);
#else
    asm volatile("s_wait_tensorcnt 0x0" ::: "memory");
#endif
}

__device__ __forceinline__ void wait_asynccnt0() {
#if defined(__AMDGCN__) && __has_builtin(__builtin_amdgcn_s_wait_asynccnt)
    __builtin_amdgcn_s_wait_asynccnt(0);
#else
    asm volatile("s_wait_asynccnt 0" ::: "memory");
#endif
}

// ---------------- pass 0: zero scratch ----------------
__global__ void zero_ws(int* __restrict__ ws, int n) {
    int i = blockIdx.x * blockDim.x + threadIdx.x;
    if (i < n) ws[i] = 0;
}

// ---------------- pass 1: fused elementwise update + per-batch histogram ----
// grid: NB*256 blocks of 256 threads; 4 pixels/thread. class_probs tile (8 x
// 1024 px = 32 KB) is DMA'd into LDS by the Tensor Data Mover (one descriptor
// per block, issued by wave 0 only), overlapping with LDS-histogram init.
__global__ __launch_bounds__(256) void fused_update_hist(
    const float* __restrict__ cur,  const float* __restrict__ cp,
    const float* __restrict__ bgp,  const float* __restrict__ pmem,
    const float* __restrict__ psc,  const float* __restrict__ cw,
    float* __restrict__ out,        int* __restrict__ hist)
{
    __shared__ int   lhist[BINS];
    __shared__ float lw[NCLS];
#if USE_TDM
    __shared__ float cptile[NCLS * TILE];      // 32 KB staged class_probs tile
#endif
    const int tid   = threadIdx.x;
    const int batch = blockIdx.x >> 8;         // 256 blocks per batch image
    const int p0    = (blockIdx.x & 255) * TILE;
    const int p     = p0 + tid * 4;
    const size_t g  = (size_t)batch * HW + p;  // [B,1,H,W] flat index

#if USE_TDM
    // ---- TDM: one wave issues the 2D tile DMA (TDM ignores EXEC, so the
    // guard must be a *scalar* branch to avoid 8 duplicate issues) ----
    if (__builtin_amdgcn_readfirstlane(tid) == 0) {
        uint64_t ga = (uint64_t)(uintptr_t)(cp + (size_t)batch * NCLS * HW + p0);
        uint32_t la = (uint32_t)(uintptr_t)(&cptile[0]);   // LDS byte address
        u32x4 g0 = { 1u,                                   // count=1 (valid D#)
                     la,                                   // lds_addr
                     (uint32_t)(ga & 0xFFFFFFFFu),         // global_addr[31:0]
                     (uint32_t)((ga >> 32) & 0x1FFFFFFu) | (2u << 30) }; // [56:32] | type=2
        i32x8 g1 = { (int)(2u << 16),                      // data_size = 4 bytes
                     0,                                    // tensor_dim0[15:0] (0x40000 lo16)
                     (int)((8u << 16) | 4u),               // tensor_dim1=8 | tensor_dim0 hi16
                     (int)(((unsigned)TILE) << 16),        // tile_dim0 = 1024
                     8,                                    // tile_dim1 = 8 classes
                     (int)0x40000,                         // tensor_dim0_stride = HW (lo32)
                     0, 0 };                               // stride hi / dim1_stride unused
        i32x4 g2 = { 1, 1, 0, 0 };                         // tensor_dim2=1, tensor_dim3=1
        i32x4 g3 = { 0, 1 << 16, 0, 0 };                   // tensor_dim4=1, tile_dim3/4=0
#if __clang_major__ >= 23
        i32x8 g5 = { 0, 0, 0, 0, 0, 0, 0, 0 };
        __builtin_amdgcn_tensor_load_to_lds(g0, g1, g2, g3, g5, 0);
#else
        __builtin_amdgcn_tensor_load_to_lds(g0, g1, g2, g3, 0);
#endif
    }
#endif

    for (int i = tid; i < BINS; i += 256) lhist[i] = 0;

    // async global->LDS broadcast of the 8 class weights (ASYNCcnt path)
    if (tid < NCLS) {
        uint32_t lds_addr = (uint32_t)(uintptr_t)(&lw[tid]);
        uint64_t gaddr    = (uint64_t)(uintptr_t)(cw + tid);
        asm volatile("global_load_async_to_lds_b32 %0, %1, off"
                     :: "v"(lds_addr), "v"(gaddr) : "memory");
    }
    wait_asynccnt0();
    wait_tensorcnt0();       // waves with no TDM issued have TENSORcnt==0: no-op
    __syncthreads();

    float w[NCLS];
#pragma unroll
    for (int c = 0; c < NCLS; ++c) w[c] = lw[c];

    float imp[4] = {0.f, 0.f, 0.f, 0.f};
    float mxc[4] = {-1e30f, -1e30f, -1e30f, -1e30f};
#pragma unroll
    for (int c = 0; c < NCLS; ++c) {
#if USE_TDM
        float4 v = *reinterpret_cast<const float4*>(&cptile[c * TILE + tid * 4]);
#else
        float4 v = *reinterpret_cast<const float4*>(cp + ((size_t)(batch * NCLS + c)) * HW + p);
#endif
        float a[4] = {v.x, v.y, v.z, v.w};
#pragma unroll
        for (int j = 0; j < 4; ++j) {
            imp[j] = fmaf(a[j], w[c], imp[j]);
            mxc[j] = fmaxf(mxc[j], a[j]);
        }
    }
    float4 b4 = *reinterpret_cast<const float4*>(bgp + g);
    float4 s4 = *reinterpret_cast<const float4*>(psc + g);
    float bg[4] = {b4.x, b4.y, b4.z, b4.w};
    float ps[4] = {s4.x, s4.y, s4.z, s4.w};

    float sc[4], im[4], wmf[4];
    int wm[4];
#pragma unroll
    for (int j = 0; j < 4; ++j) {
        // argmax([bg, classes]) != 0  <=>  max(classes) > bg  (ties -> background)
        float fg  = (mxc[j] > bg[j]) ? imp[j] : 0.0f;
        float dec = ps[j] * DECAY;
        wm[j]  = fg > dec + TAU;
        sc[j]  = wm[j] ? fg : dec;
        im[j]  = fg;
        wmf[j] = wm[j] ? 1.0f : 0.0f;
        atomicAdd(&lhist[bin_of(sc[j])], 1);          // ds_add_u32
    }
    *reinterpret_cast<float4*>(out + OFF_SCORE + g) = make_float4(sc[0], sc[1], sc[2], sc[3]);
    *reinterpret_cast<float4*>(out + OFF_IMP   + g) = make_float4(im[0], im[1], im[2], im[3]);
    *reinterpret_cast<float4*>(out + OFF_WMASK + g) = make_float4(wmf[0], wmf[1], wmf[2], wmf[3]);

#pragma unroll
    for (int ch = 0; ch < NC; ++ch) {
        size_t mi = ((size_t)(batch * NC + ch)) * HW + p;
        float4 c4 = *reinterpret_cast<const float4*>(cur  + mi);
        float4 m4 = *reinterpret_cast<const float4*>(pmem + mi);
        float4 r;
        r.x = wm[0] ? c4.x : m4.x;
        r.y = wm[1] ? c4.y : m4.y;
        r.z = wm[2] ? c4.z : m4.z;
        r.w = wm[3] ? c4.w : m4.w;
        *reinterpret_cast<float4*>(out + OFF_MEMST + mi) = r;
    }

    __syncthreads();
    int* gh = hist + batch * BINS;
    for (int i = tid; i < BINS; i += 256) {
        int v = lhist[i];
        if (v) atomicAdd(&gh[i], v);                  // global_atomic_add_u32
    }
}

// ---------------- pass 2: per-batch threshold bin + tie quota ----------------
__global__ void find_threshold(const int* __restrict__ hist,
                               int* __restrict__ thr, int* __restrict__ quo)
{
    const int b = blockIdx.x;
    if (threadIdx.x != 0) return;
    const int* h = hist + b * BINS;
    int cum = 0, t = 0, q = KEEP;
    for (int i = BINS - 1; i >= 0; --i) {
        int c = h[i];
        if (cum + c >= KEEP) { t = i; q = KEEP - cum; break; }
        cum += c;
    }
    thr[b] = t;
    quo[b] = q;
}

// ---------------- pass 3: per-block count of tie-bin pixels ----------------
__global__ __launch_bounds__(256) void count_tie(const float* __restrict__ score,
                                                 const int* __restrict__ thr,
                                                 int* __restrict__ bcnt)
{
    __shared__ int cnt;
    const int tid   = threadIdx.x;
    const int batch = blockIdx.x >> 8;
    const int blk   = blockIdx.x & 255;
    if (tid == 0) cnt = 0;
    __syncthreads();
    const int    t = thr[batch];
    const size_t g = (size_t)batch * HW + blk * TILE + tid * 4;
    float4 s4 = *reinterpret_cast<const float4*>(score + g);
    float s[4] = {s4.x, s4.y, s4.z, s4.w};
    int local = 0;
#pragma unroll
    for (int j = 0; j < 4; ++j) local += (bin_of(s[j]) == t);
    if (local) atomicAdd(&cnt, local);
    __syncthreads();
    if (tid == 0) bcnt[batch * 256 + blk] = cnt;
}

// ---------------- pass 4: per-batch exclusive scan of block counts --------
__global__ __launch_bounds__(256) void scan_counts(int* __restrict__ bcnt)
{
    __shared__ int s[256];
    const int b = blockIdx.x, tid = threadIdx.x;
    const int orig = bcnt[b * 256 + tid];
    s[tid] = orig;
    __syncthreads();
    for (int off = 1; off < 256; off <<= 1) {
        int v = (tid >= off) ? s[tid - off] : 0;
        __syncthreads();
        s[tid] += v;
        __syncthreads();
    }
    bcnt[b * 256 + tid] = s[tid] - orig;   // exclusive prefix
}

// ---------------- pass 5: deterministic top-k mask + masked memory image ---
__global__ __launch_bounds__(256) void finalize_topk(
    const float* __restrict__ score, const float* __restrict__ memst,
    const int* __restrict__ thr,     const int* __restrict__ quo,
    const int* __restrict__ boff,
    float* __restrict__ omask,       float* __restrict__ memimg)
{
    __shared__ int s[256];
    const int tid   = threadIdx.x;
    const int batch = blockIdx.x >> 8;
    const int blk   = blockIdx.x & 255;
    const int t     = thr[batch];
    const int q     = quo[batch];
    const int base  = boff[batch * 256 + blk];
    const int p     = blk * TILE + tid * 4;
    const size_t g  = (size_t)batch * HW + p;

    float4 s4 = *reinterpret_cast<const float4*>(score + g);
    float sv[4] = {s4.x, s4.y, s4.z, s4.w};
    int bn[4];
    int myt = 0;
#pragma unroll
    for (int j = 0; j < 4; ++j) { bn[j] = bin_of(sv[j]); myt += (bn[j] == t); }

    s[tid] = myt;
    __syncthreads();
    for (int off = 1; off < 256; off <<= 1) {
        int v = (tid >= off) ? s[tid - off] : 0;
        __syncthreads();
        s[tid] += v;
        __syncthreads();
    }
    int rank = base + (s[tid] - myt);   // index-ordered rank among tie pixels

    float m[4];
#pragma unroll
    for (int j = 0; j < 4; ++j) {
        if (bn[j] > t)        m[j] = 1.0f;
        else if (bn[j] == t) { m[j] = (rank < q) ? 1.0f : 0.0f; rank++; }
        else                  m[j] = 0.0f;
    }
    *reinterpret_cast<float4*>(omask + g) = make_float4(m[0], m[1], m[2], m[3]);

#pragma unroll
    for (int ch = 0; ch < NC; ++ch) {
        size_t mi = ((size_t)(batch * NC + ch)) * HW + p;
        float4 ms = *reinterpret_cast<const float4*>(memst + mi);
        float4 r  = make_float4(ms.x * m[0], ms.y * m[1], ms.z * m[2], ms.w * m[3]);
        *reinterpret_cast<float4*>(memimg + mi) = r;
    }
}

extern "C" void kernel_launch(void* const* d_in, const int* in_sizes, int n_in,
                              void* d_out, int out_size, void* d_ws, size_t ws_size,
                              hipStream_t stream) {
    (void)in_sizes; (void)n_in; (void)out_size; (void)ws_size;
    const float* cur = (const float*)d_in[0];   // current_image  [16,3,512,512]
    const float* cp  = (const float*)d_in[1];   // class_probs    [16,8,512,512]
    const float* bg  = (const float*)d_in[2];   // background_prob[16,1,512,512]
    const float* pm  = (const float*)d_in[3];   // prev_memory    [16,3,512,512]
    const float* ps  = (const float*)d_in[4];   // prev_scores    [16,1,512,512]
    const float* cw  = (const float*)d_in[5];   // class_weights  [1,8,1,1]
    float* out = (float*)d_out;
    int*   ws  = (int*)d_ws;                    // needs >= ~272 KB

    const int grid = NB * 256;                  // 1024 px per block

    zero_ws<<<(WS_TOTAL + 255) / 256, 256, 0, stream>>>(ws, WS_TOTAL);
    fused_update_hist<<<grid, 256, 0, stream>>>(cur, cp, bg, pm, ps, cw,
                                                out, ws + WS_HIST);
    find_threshold<<<NB, 32, 0, stream>>>(ws + WS_HIST, ws + WS_THR, ws + WS_QUO);
    count_tie<<<grid, 256, 0, stream>>>(out + OFF_SCORE, ws + WS_THR, ws + WS_BCNT);
    scan_counts<<<NB, 256, 0, stream>>>(ws + WS_BCNT);
    finalize_topk<<<grid, 256, 0, stream>>>(out + OFF_SCORE, out + OFF_MEMST,
                                            ws + WS_THR, ws + WS_QUO, ws + WS_BCNT,
                                            out + OFF_OMASK, out + OFF_MEMIMG);
}